// USR_EMB_23562190586374
// MI455X (gfx1250) — compile-verified
//
#include <hip/hip_runtime.h>

// Problem shape (fixed by the reference).
#define EMB_DIM   64            // floats per embedding row (256 bytes)
#define Q_PER_ROW (EMB_DIM / 4) // 16 float4 per row

// Native clang vector type: accepted by __builtin_nontemporal_load/store,
// lowers to global_load_b128 / global_store_b128.
typedef float v4f __attribute__((ext_vector_type(4)));

// ---------------------------------------------------------------------------
// Kernel 1: invert the userlist permutation.
//   userlist is a permutation of 0..n-1 (entry 0 == 0), so every slot of
//   inv[] is written each call -> deterministic, no init needed.
// ---------------------------------------------------------------------------
__global__ __launch_bounds__(256) void usr_emb_build_inverse(
    const int* __restrict__ userlist,
    int* __restrict__ inv,
    int n) {
  int i = blockIdx.x * blockDim.x + threadIdx.x;
  if (i < n) {
    inv[userlist[i]] = i;  // scatter: inv[value] = position
  }
}

// ---------------------------------------------------------------------------
// Kernel 2: gather embedding rows.
//   One thread moves one float4 (16 B). 16 threads cover one 256-B row, so a
//   wave32 streams 512 B fully coalesced (global_load_b128 / store_b128).
//   - x is read once  -> non-temporal load
//   - out is written once and is 210 MB -> non-temporal store so the 25.6 MB
//     emb table stays resident in the 192 MB L2 (8x average row reuse).
// ---------------------------------------------------------------------------
__global__ __launch_bounds__(256) void usr_emb_gather(
    const int* __restrict__ x,        // [nrows] flattened (B*L)
    const int* __restrict__ inv,      // [usr_size] inverse permutation
    const v4f* __restrict__ emb4,     // [usr_size * Q_PER_ROW]
    v4f* __restrict__ out4,           // [nrows   * Q_PER_ROW]
    int nrows) {
  int t   = blockIdx.x * blockDim.x + threadIdx.x;
  int row = t >> 4;                 // which (b, l) pair
  if (row >= nrows) return;
  int q   = t & 15;                 // which float4 within the row

  int xv = __builtin_nontemporal_load(x + row);   // single-use index stream
  int id = inv[xv];                               // invert permutation

  v4f v = emb4[(size_t)id * Q_PER_ROW + q];       // cached: emb lives in L2
  __builtin_nontemporal_store(v, out4 + (size_t)row * Q_PER_ROW + q);
}

// ---------------------------------------------------------------------------
// Launch: inputs in setup_inputs() order: x [B*L], userlist [usr_size],
// emb [usr_size*EMB_DIM]. Output: float32 [B*L*EMB_DIM]. d_ws holds inv[].
// ---------------------------------------------------------------------------
extern "C" void kernel_launch(void* const* d_in, const int* in_sizes, int n_in,
                              void* d_out, int out_size, void* d_ws, size_t ws_size,
                              hipStream_t stream) {
  const int*   x        = (const int*)d_in[0];
  const int*   userlist = (const int*)d_in[1];
  const float* emb      = (const float*)d_in[2];
  float*       out      = (float*)d_out;
  int*         inv      = (int*)d_ws;            // usr_size * 4 bytes scratch

  const int nrows    = in_sizes[0];              // B*L = 819200
  const int usr_size = in_sizes[1];              // 100001

  // 1) Build inverse permutation (fully overwrites inv each call).
  {
    int blocks = (usr_size + 255) / 256;
    usr_emb_build_inverse<<<blocks, 256, 0, stream>>>(userlist, inv, usr_size);
  }

  // 2) Gather: one thread per float4 of output.
  {
    long long nquads = (long long)nrows * Q_PER_ROW;  // 13,107,200
    int blocks = (int)((nquads + 255) / 256);
    usr_emb_gather<<<blocks, 256, 0, stream>>>(
        x, inv, (const v4f*)emb, (v4f*)out, nrows);
  }
}